// ReentrancyDetectionHead_81587198755028
// MI455X (gfx1250) — compile-verified
//
#include <hip/hip_runtime.h>
#include <hip/hip_bf16.h>
#include <math.h>

typedef __attribute__((ext_vector_type(2))) float v2f;
typedef __attribute__((ext_vector_type(4))) float v4f;
typedef __attribute__((ext_vector_type(8))) float v8f;

#define B_GRAPHS 4096
#define HIDDEN   256

// ---------------------------------------------------------------------------
// Kernel 1: per-graph segment mean of columns (8,10,14,15).
// batch is sorted -> one block per graph, binary search the node range,
// strided block reduction, no atomics, exact count from the range.
//
// node_features (~1 GB) is streamed exactly once -> non-temporal loads so the
// stream does not evict batch (8 MB, reused by 4096 binary searches) from the
// 192 MB L2. ws layout: float means[B][4]; float counts[B]
// ---------------------------------------------------------------------------
__global__ __launch_bounds__(256)
void seg_mean_kernel(const float* __restrict__ nf,
                     const long long* __restrict__ batch,
                     float* __restrict__ means,
                     float* __restrict__ counts,
                     int n_nodes)
{
    const int b   = blockIdx.x;      // graph id 0..4095
    const int tid = threadIdx.x;

    __shared__ int s_range[2];
    if (tid == 0) {
        // lower_bound(batch, b)
        int lo = 0, hi = n_nodes;
        while (lo < hi) { int mid = (lo + hi) >> 1; if (batch[mid] < (long long)b) lo = mid + 1; else hi = mid; }
        s_range[0] = lo;
        // lower_bound(batch, b+1), resume from previous result
        hi = n_nodes;
        while (lo < hi) { int mid = (lo + hi) >> 1; if (batch[mid] < (long long)(b + 1)) lo = mid + 1; else hi = mid; }
        s_range[1] = lo;
    }
    __syncthreads();
    const int start = s_range[0];
    const int end   = s_range[1];

    float a0 = 0.f, a1 = 0.f, a2 = 0.f, a3 = 0.f;
    for (int i = start + tid; i < end; i += 256) {
        const float* row = nf + (size_t)i * HIDDEN;
        // speculative prefetch of the next row this thread will touch
        __builtin_prefetch(row + 256 * HIDDEN + 8, 0, 0);
        // streaming (non-temporal) loads: cols 8..11 (16B aligned), 14..15 (8B)
        v4f c8  = __builtin_nontemporal_load((const v4f*)(row + 8));
        v2f c14 = __builtin_nontemporal_load((const v2f*)(row + 14));
        a0 += c8.x;   // col 8
        a1 += c8.z;   // col 10
        a2 += c14.x;  // col 14
        a3 += c14.y;  // col 15
    }

    __shared__ float red[4][256];
    red[0][tid] = a0; red[1][tid] = a1; red[2][tid] = a2; red[3][tid] = a3;
    __syncthreads();
    for (int s = 128; s > 0; s >>= 1) {
        if (tid < s) {
            red[0][tid] += red[0][tid + s];
            red[1][tid] += red[1][tid + s];
            red[2][tid] += red[2][tid + s];
            red[3][tid] += red[3][tid + s];
        }
        __syncthreads();
    }
    if (tid == 0) {
        const float cnt = (float)(end - start);
        const float inv = 1.0f / fmaxf(cnt, 1.0f);
        means[b * 4 + 0] = red[0][0] * inv;
        means[b * 4 + 1] = red[1][0] * inv;
        means[b * 4 + 2] = red[2][0] * inv;
        means[b * 4 + 3] = red[3][0] * inv;
        counts[b] = cnt;
    }
}

// ---------------------------------------------------------------------------
// Kernel 2: MLP 4->64 (relu) ->32 (relu) ->1 (sigmoid) on 4096 rows using
// V_WMMA_F32_16X16X4_F32. One wave per 16-row tile; 4 waves per block.
//
// f32 WMMA operand layouts (ISA 7.12.2, wave32):
//   A (16x4):  lane L -> row M = L&15; VGPR j, group g=L>>4 -> K = j + 2g
//   B (4x16):  lane L -> col N = L&15; VGPR j, group g      -> K = j + 2g
//   C/D(16x16):lane L -> col N = L&15; VGPR v, group g      -> row M = v + 8g
// ---------------------------------------------------------------------------
__global__ __launch_bounds__(128)
void mlp_kernel(const float* __restrict__ means,
                const float* __restrict__ counts,
                const float* __restrict__ W1, const float* __restrict__ b1,
                const float* __restrict__ W2, const float* __restrict__ b2,
                const float* __restrict__ W3, const float* __restrict__ b3,
                float* __restrict__ out)
{
    const int tid  = threadIdx.x;
    const int wave = tid >> 5;
    const int lane = tid & 31;
    const int tile = blockIdx.x * 4 + wave;     // 0..255 (4096/16 tiles)
    const int m    = lane & 15;
    const int g    = lane >> 4;
    const int k0   = g * 2;
    const int row  = tile * 16 + m;

    __shared__ float s_h1[4][16][64];   // per-wave layer-1 activations
    __shared__ float s_h2[4][16][32];   // per-wave layer-2 activations

    // ---- Layer 1: (16x4) @ (4x64), 4 column tiles of 16 --------------------
    // A operand: means[row][k0], means[row][k0+1] (8B aligned -> b64 load)
    v2f av1 = *(const v2f*)(means + row * 4 + k0);

    for (int nt = 0; nt < 4; ++nt) {
        const int col = nt * 16 + m;            // B/D column for this lane
        v2f bv;
        bv.x = W1[k0       * 64 + col];
        bv.y = W1[(k0 + 1) * 64 + col];
        v8f c = {};
        c = __builtin_amdgcn_wmma_f32_16x16x4_f32(
                false, av1, false, bv, (short)0, c, false, false);
        const float bias = b1[col];
#pragma unroll
        for (int v = 0; v < 8; ++v) {
            const int r = v + 8 * g;
            s_h1[wave][r][col] = fmaxf(c[v] + bias, 0.0f);
        }
    }
    __syncthreads();

    // ---- Layer 2: (16x64) @ (64x32), K chained as 16 x (K=4) ---------------
    for (int nt = 0; nt < 2; ++nt) {
        const int col = nt * 16 + m;
        v8f c = {};
#pragma unroll
        for (int kk = 0; kk < 16; ++kk) {
            const int kb = kk * 4 + k0;         // even -> 8B aligned LDS read
            v2f avx = *(const v2f*)&s_h1[wave][m][kb];
            v2f bvx;
            bvx.x = W2[kb       * 32 + col];
            bvx.y = W2[(kb + 1) * 32 + col];
            c = __builtin_amdgcn_wmma_f32_16x16x4_f32(
                    false, avx, false, bvx, (short)0, c, false, false);
        }
        const float bias = b2[col];
#pragma unroll
        for (int v = 0; v < 8; ++v) {
            const int r = v + 8 * g;
            s_h2[wave][r][col] = fmaxf(c[v] + bias, 0.0f);
        }
    }
    __syncthreads();

    // ---- Layer 3: (16x32) @ (32x1) + sigmoid, lanes 0..15 ------------------
    if (g == 0) {
        float acc = b3[0];
#pragma unroll
        for (int k = 0; k < 32; ++k)
            acc += s_h2[wave][m][k] * W3[k];
        const float score = 1.0f / (1.0f + expf(-acc));
        const float cnt   = counts[row];
        out[row] = (cnt > 0.0f) ? score : 0.0f;
    }
}

// ---------------------------------------------------------------------------
extern "C" void kernel_launch(void* const* d_in, const int* in_sizes, int n_in,
                              void* d_out, int out_size, void* d_ws, size_t ws_size,
                              hipStream_t stream)
{
    const float*     nf    = (const float*)d_in[0];
    const long long* batch = (const long long*)d_in[1];
    // d_in[2] = graph_embedding (unused by the reference computation)
    const float* W1 = (const float*)d_in[3];
    const float* b1 = (const float*)d_in[4];
    const float* W2 = (const float*)d_in[5];
    const float* b2 = (const float*)d_in[6];
    const float* W3 = (const float*)d_in[7];
    const float* b3 = (const float*)d_in[8];
    float* out = (float*)d_out;

    const int n_nodes = in_sizes[1];

    float* means  = (float*)d_ws;             // [4096][4]
    float* counts = means + B_GRAPHS * 4;     // [4096]

    seg_mean_kernel<<<B_GRAPHS, 256, 0, stream>>>(nf, batch, means, counts, n_nodes);

    // 4096 rows / 16 rows-per-wave / 4 waves-per-block = 64 blocks
    mlp_kernel<<<B_GRAPHS / 16 / 4, 128, 0, stream>>>(means, counts,
                                                      W1, b1, W2, b2, W3, b3, out);
}